// BesSimpleTransformer_56470230008152
// MI455X (gfx1250) — compile-verified
//
#include <hip/hip_runtime.h>
#include <hip/hip_bf16.h>

// Tiny transformer (S=26, D=7, DA=11, V=29) on gfx1250 (MI455X).
// Latency-bound problem (~56 KFLOP, ~3 KB in / ~6 KB out): single block,
// single wave32, all intermediates in LDS, matmuls via V_WMMA_F32_16X16X4_F32
// (exact f32 RNE semantics) on zero-padded 16x16 tiles. Biases staged in
// zero-padded LDS so tile stores are branch-free; block loop unrolled with
// concrete pointers so weight staging uses direct global loads (no flat, no
// per-use s_cselect); staging loops templated/unrolled (no runtime division).

static constexpr int SEQ  = 26;  // sequence length
static constexpr int EMB  = 7;   // embedding dim
static constexpr int ADIM = 11;  // attention dim
static constexpr int VOC  = 29;  // vocab

typedef __attribute__((ext_vector_type(2))) float v2f;
typedef __attribute__((ext_vector_type(8))) float v8f;

// C[16x16] = sum_k A[16xKpad] * B[Kpadx16], A/B row-major in LDS, Kpad%4==0.
// A-frag: lane<16 -> K={k0,k0+1}, lane>=16 -> K={k0+2,k0+3}, row = lane&15.
// B-frag mirrors A with col = lane&15.
__device__ __forceinline__ v8f wmma_tile(const float* A, int lda, int mo,
                                         const float* B, int ldb, int no,
                                         int kpad) {
  const int lane = threadIdx.x & 31;
  const int half = lane >> 4;   // 0 or 1
  const int idx  = lane & 15;   // A row / B col within tile
  v8f c = {0.f, 0.f, 0.f, 0.f, 0.f, 0.f, 0.f, 0.f};
#pragma unroll
  for (int k0 = 0; k0 < kpad; k0 += 4) {
    const int kb = k0 + half * 2;
    v2f a, b;
    a.x = A[(mo + idx) * lda + kb + 0];
    a.y = A[(mo + idx) * lda + kb + 1];
    b.x = B[(kb + 0) * ldb + no + idx];
    b.y = B[(kb + 1) * ldb + no + idx];
    c = __builtin_amdgcn_wmma_f32_16x16x4_f32(
        /*neg_a=*/false, a, /*neg_b=*/false, b,
        /*c_mod=*/(short)0, c, /*reuse_a=*/false, /*reuse_b=*/false);
  }
  return c;
}

// Write one 16x16 D-tile to LDS. biasLds is a zero-padded LDS vector.
// Columns beyond the valid width are already exactly 0 in c (zero-padded B)
// and 0 in biasLds, so only the row mask is needed -> v_cndmask codegen.
__device__ __forceinline__ void store_tile(float* C, int ldc, int mo, int no,
                                           v8f c, int mvalid,
                                           const float* biasLds) {
  const int lane = threadIdx.x & 31;
  const int half = lane >> 4;
  const int n = (lane & 15) + no;
  const float b = biasLds[n];
#pragma unroll
  for (int v = 0; v < 8; ++v) {
    const int m = mo + v + half * 8;
    C[m * ldc + n] = (m < mvalid) ? (c[v] + b) : 0.f;
  }
}

// WT[r][c] = w[c*INF + r] (16 rows x LDW cols, zero padded). Compile-time
// shapes: contiguous coalesced global read, fully unrolled, constant-divisor
// index math, single tail predicate.
template <int OUTF, int INF, int LDW>
__device__ __forceinline__ void fill_wT(float* WT, const float* __restrict__ w) {
  const int lane = threadIdx.x & 31;
#pragma unroll
  for (int i0 = 0; i0 < 16 * LDW; i0 += 32) WT[i0 + lane] = 0.f;
  __syncthreads();
#pragma unroll
  for (int i0 = 0; i0 < OUTF * INF; i0 += 32) {
    const int i = i0 + lane;
    if (i < OUTF * INF) {
      const int o = i / INF;        // constant divisor -> mul-hi
      const int r = i - o * INF;
      WT[r * LDW + o] = w[i];
    }
  }
}

// One transformer block, fully specialized on concrete weight pointers so all
// weight/bias loads are direct global loads (no flat, no pointer selects).
__device__ __forceinline__ void run_block(
    float* H, float* Kl, float* Ql, float* Vl, float* KT, float* ATT,
    float* RES, float* WT,
    float* BKl, float* BQl, float* BVl, float* BFl, const float* ZB,
    const float* __restrict__ wk, const float* __restrict__ bk,
    const float* __restrict__ wq, const float* __restrict__ bq,
    const float* __restrict__ wv, const float* __restrict__ bv,
    const float* __restrict__ wf, const float* __restrict__ bf) {
  const int lane = threadIdx.x & 31;

  // ---- Stage this block's biases (zero padded) ----
  if (lane < 16) {
    BKl[lane] = (lane < ADIM) ? bk[lane] : 0.f;
    BQl[lane] = (lane < ADIM) ? bq[lane] : 0.f;
    BVl[lane] = (lane < ADIM) ? bv[lane] : 0.f;
    BFl[lane] = (lane < EMB)  ? bf[lane] : 0.f;
  }
  __syncthreads();

  // ---- K = H @ wk^T + bk ----
  fill_wT<ADIM, EMB, 16>(WT, wk);
  __syncthreads();
  for (int mo = 0; mo < 32; mo += 16)
    store_tile(Kl, 16, mo, 0, wmma_tile(H, 16, mo, WT, 16, 0, 8), SEQ, BKl);
  __syncthreads();

  // ---- Q = H @ wq^T + bq ----
  fill_wT<ADIM, EMB, 16>(WT, wq);
  __syncthreads();
  for (int mo = 0; mo < 32; mo += 16)
    store_tile(Ql, 16, mo, 0, wmma_tile(H, 16, mo, WT, 16, 0, 8), SEQ, BQl);
  __syncthreads();

  // ---- V = H @ wv^T + bv ----
  fill_wT<ADIM, EMB, 16>(WT, wv);
  __syncthreads();
  for (int mo = 0; mo < 32; mo += 16)
    store_tile(Vl, 16, mo, 0, wmma_tile(H, 16, mo, WT, 16, 0, 8), SEQ, BVl);
  __syncthreads();

  // ---- KT[d][j] = Kl[j][d]  (16x32; shift/mask indexing, unrolled) ----
#pragma unroll
  for (int i0 = 0; i0 < 16 * 32; i0 += 32) {
    const int i = i0 + lane;
    KT[i] = Kl[(i & 31) * 16 + (i >> 5)];
  }
  __syncthreads();

  // ---- ATT = Q @ K^T (raw logits; padded region exactly zero) ----
  for (int mo = 0; mo < 32; mo += 16)
    for (int no = 0; no < 32; no += 16)
      store_tile(ATT, 32, mo, no, wmma_tile(Ql, 16, mo, KT, 32, no, 16),
                 SEQ, ZB);
  __syncthreads();

  // ---- Causal softmax, one lane per row; pads written to 0 ----
  {
    const int i = lane;
    if (i < SEQ) {
      float mx = ATT[i * 32];
      for (int j = 1; j <= i; ++j) mx = fmaxf(mx, ATT[i * 32 + j]);
      float s = 0.f;
      for (int j = 0; j <= i; ++j) s += __expf(ATT[i * 32 + j] - mx);
      const float inv = 1.0f / s;
      for (int j = 0; j < 32; ++j)
        ATT[i * 32 + j] = (j <= i) ? __expf(ATT[i * 32 + j] - mx) * inv : 0.f;
    } else {
      for (int j = 0; j < 32; ++j) ATT[i * 32 + j] = 0.f;
    }
  }
  __syncthreads();

  // ---- RES = ATT @ V  (K-dim 32, pads are zero) ----
  for (int mo = 0; mo < 32; mo += 16)
    store_tile(RES, 16, mo, 0, wmma_tile(ATT, 32, mo, Vl, 16, 0, 32),
               SEQ, ZB);
  __syncthreads();

  // ---- H = RES @ wf^T + bf ----
  fill_wT<EMB, ADIM, 16>(WT, wf);
  __syncthreads();
  for (int mo = 0; mo < 32; mo += 16)
    store_tile(H, 16, mo, 0, wmma_tile(RES, 16, mo, WT, 16, 0, 16),
               SEQ, BFl);
  __syncthreads();
}

__global__ __launch_bounds__(32)
void bes_transformer_kernel(const int* __restrict__ x,
                            const float* __restrict__ emb_table,
                            const float* __restrict__ pos,
                            const float* __restrict__ wk0, const float* __restrict__ bk0,
                            const float* __restrict__ wq0, const float* __restrict__ bq0,
                            const float* __restrict__ wv0, const float* __restrict__ bv0,
                            const float* __restrict__ wf0, const float* __restrict__ bf0,
                            const float* __restrict__ wk1, const float* __restrict__ bk1,
                            const float* __restrict__ wq1, const float* __restrict__ bq1,
                            const float* __restrict__ wv1, const float* __restrict__ bv1,
                            const float* __restrict__ wf1, const float* __restrict__ bf1,
                            const float* __restrict__ wout, const float* __restrict__ bout,
                            float* __restrict__ out) {
  const int lane = threadIdx.x & 31;

  __shared__ float H[32 * 16];     // hidden state, padded
  __shared__ float Kl[32 * 16];    // K projection
  __shared__ float Ql[32 * 16];    // Q projection
  __shared__ float Vl[32 * 16];    // V projection
  __shared__ float KT[16 * 32];    // K^T for att = Q @ K^T
  __shared__ float ATT[32 * 32];   // attention matrix
  __shared__ float RES[32 * 16];   // att @ V
  __shared__ float WT[16 * 16];    // staged transposed weight (reused)
  __shared__ float WOT[16 * 32];   // w_out^T (16x32 padded)
  __shared__ float BKl[16], BQl[16], BVl[16], BFl[16];
  __shared__ float BOl[32];        // b_out padded to 32
  __shared__ float ZB[32];         // all-zero "bias"

  // ---- One-time staging ----
#pragma unroll
  for (int i0 = 0; i0 < 32 * 16; i0 += 32) H[i0 + lane] = 0.f;
  ZB[lane]  = 0.f;
  BOl[lane] = (lane < VOC) ? bout[lane] : 0.f;
  __syncthreads();

  // ---- Embedding: H = emb_table[x] + pos ----
  if (lane < SEQ) {
    const int tok = x[lane];
#pragma unroll
    for (int d = 0; d < EMB; ++d)
      H[lane * 16 + d] = emb_table[tok * EMB + d] + pos[lane * EMB + d];
  }
  __syncthreads();

  // ---- Two transformer blocks (unrolled: concrete pointers) ----
  run_block(H, Kl, Ql, Vl, KT, ATT, RES, WT, BKl, BQl, BVl, BFl, ZB,
            wk0, bk0, wq0, bq0, wv0, bv0, wf0, bf0);
  run_block(H, Kl, Ql, Vl, KT, ATT, RES, WT, BKl, BQl, BVl, BFl, ZB,
            wk1, bk1, wq1, bq1, wv1, bv1, wf1, bf1);

  // ---- out = H @ w_out^T + b_out  -> d_out[0 .. 26*29) ----
  fill_wT<VOC, EMB, 32>(WOT, wout);
  __syncthreads();
  {
    const int half = lane >> 4;
    const int ncol = lane & 15;
    for (int mo = 0; mo < 32; mo += 16)
      for (int no = 0; no < 32; no += 16) {
        v8f c = wmma_tile(H, 16, mo, WOT, 32, no, 8);
        const int n = ncol + no;
        const float b = BOl[n];
#pragma unroll
        for (int v = 0; v < 8; ++v) {
          const int m = mo + v + half * 8;
          if (m < SEQ && n < VOC) out[m * VOC + n] = c[v] + b;
        }
      }
  }

  // ---- att (block-2 softmax) -> d_out[26*29 .. 26*29 + 26*26) ----
  float* att_out = out + SEQ * VOC;
  for (int i = lane; i < SEQ * SEQ; i += 32) {
    const int r = i / SEQ, cc = i - r * SEQ;
    att_out[i] = ATT[r * 32 + cc];
  }
}

extern "C" void kernel_launch(void* const* d_in, const int* in_sizes, int n_in,
                              void* d_out, int out_size, void* d_ws, size_t ws_size,
                              hipStream_t stream) {
  (void)in_sizes; (void)n_in; (void)out_size; (void)d_ws; (void)ws_size;
  bes_transformer_kernel<<<1, 32, 0, stream>>>(
      (const int*)d_in[0],          // x
      (const float*)d_in[1],        // emb_table
      (const float*)d_in[2],        // pos
      (const float*)d_in[3],  (const float*)d_in[4],   // w_k0, b_k0
      (const float*)d_in[5],  (const float*)d_in[6],   // w_q0, b_q0
      (const float*)d_in[7],  (const float*)d_in[8],   // w_v0, b_v0
      (const float*)d_in[9],  (const float*)d_in[10],  // w_f0, b_f0
      (const float*)d_in[11], (const float*)d_in[12],  // w_k1, b_k1
      (const float*)d_in[13], (const float*)d_in[14],  // w_q1, b_q1
      (const float*)d_in[15], (const float*)d_in[16],  // w_v1, b_v1
      (const float*)d_in[17], (const float*)d_in[18],  // w_f1, b_f1
      (const float*)d_in[19], (const float*)d_in[20],  // w_out, b_out
      (float*)d_out);
}